// AttentionBlock_25692494365073
// MI455X (gfx1250) — compile-verified
//
#include <hip/hip_runtime.h>
#include <math.h>

#define CCH 64
#define HID 4

typedef __attribute__((ext_vector_type(2))) float v2f;
typedef __attribute__((ext_vector_type(8))) float v8f;

// ---------------------------------------------------------------- utilities
__device__ __forceinline__ float sigmoidf_(float x) {
  return 1.0f / (1.0f + expf(-x));
}

__device__ __forceinline__ void atomicMaxF(float* addr, float val) {
  int* ai = (int*)addr;
  int cur = __float_as_int(*addr);
  while (val > __int_as_float(cur)) {
    int old = atomicCAS(ai, cur, __float_as_int(val));
    if (old == cur) break;
    cur = old;
  }
}

// ---------------------------------------------------------------- 1) init
__global__ void k_init(float* __restrict__ deg, float* __restrict__ sumbuf,
                       float* __restrict__ maxbuf, int n) {
  int i = blockIdx.x * blockDim.x + threadIdx.x;
  if (i < n) deg[i] = 1.0f;  // self-loop contribution
  if (i < CCH) {
    sumbuf[i] = 0.0f;
    maxbuf[i] = -INFINITY;
  }
}

// ---------------------------------------------------------------- 2) degree
__global__ void k_degree(const int* __restrict__ dst, float* __restrict__ deg,
                         int nE) {
  int e = blockIdx.x * blockDim.x + threadIdx.x;
  if (e < nE) atomicAdd(&deg[dst[e]], 1.0f);
}

// ---------------------------------------------------------------- 3) dinv
__global__ void k_dinv(float* __restrict__ deg, int n) {
  int i = blockIdx.x * blockDim.x + threadIdx.x;
  if (i < n) {
    float d = deg[i];
    deg[i] = (d > 0.0f) ? rsqrtf(d) : 0.0f;  // in-place: deg -> dinv
  }
}

// ---------------------------------------------------------------- 4) GEMM h = x @ W  (f32 WMMA 16x16x4)
// One wave computes a 16-row x 64-col tile of h; 8 waves per block -> 128 rows.
// W is staged into LDS pre-swizzled as [k-pair][col] -> float2(W[2p][c], W[2p+1][c])
// so every B fragment is a single aligned ds_load_b64. A fragments use a
// clamped row index so loads are unconditional b64 (no EXEC-divergent paths);
// out-of-range lanes produce garbage that the bounds-checked store discards.
__global__ void __launch_bounds__(256)
k_gemm(const float* __restrict__ x, const float* __restrict__ W,
       float* __restrict__ h, int nrows) {
  __shared__ v2f Wls[32 * CCH];  // 16 KB
  for (int i = threadIdx.x; i < 32 * CCH; i += 256) {
    int p = i >> 6;        // k-pair
    int c = i & 63;        // column
    v2f w;
    w.x = W[(2 * p + 0) * CCH + c];
    w.y = W[(2 * p + 1) * CCH + c];
    Wls[i] = w;
  }
  __syncthreads();

  const int wave = threadIdx.x >> 5;  // 0..7
  const int lane = threadIdx.x & 31;
  const int row0 = (blockIdx.x * 8 + wave) * 16;
  if (row0 >= nrows) return;          // uniform per wave

  const int half = lane >> 4;         // 0: K pair {k,k+1}, 1: {k+2,k+3}
  const int l16  = lane & 15;
  const int arow = min(row0 + l16, nrows - 1);  // clamp (no divergence)
  const float* xr = x + (size_t)arow * CCH;

  v8f acc0 = {}, acc1 = {}, acc2 = {}, acc3 = {};

#pragma unroll
  for (int k = 0; k < CCH; k += 4) {
    const int p = (k >> 1) + half;    // this lane's k-pair index
    // A fragment: ISA 32-bit A 16x4 layout (lanes 0-15: K=k,k+1; 16-31: K=k+2,k+3)
    const v2f a = *(const v2f*)(xr + 2 * p);
    // B fragments for the four 16-wide N tiles (one ds_load_b64 each)
    const v2f b0 = Wls[p * CCH +  0 + l16];
    const v2f b1 = Wls[p * CCH + 16 + l16];
    const v2f b2 = Wls[p * CCH + 32 + l16];
    const v2f b3 = Wls[p * CCH + 48 + l16];

    acc0 = __builtin_amdgcn_wmma_f32_16x16x4_f32(false, a, false, b0, (short)0, acc0, false, false);
    acc1 = __builtin_amdgcn_wmma_f32_16x16x4_f32(false, a, false, b1, (short)0, acc1, false, false);
    acc2 = __builtin_amdgcn_wmma_f32_16x16x4_f32(false, a, false, b2, (short)0, acc2, false, false);
    acc3 = __builtin_amdgcn_wmma_f32_16x16x4_f32(false, a, false, b3, (short)0, acc3, false, false);
  }

  // D layout: VGPR j -> row (j + 8*half), col = l16 (+ N-tile offset)
#pragma unroll
  for (int j = 0; j < 8; ++j) {
    const int row = row0 + j + 8 * half;
    if (row < nrows) {
      float* hr = h + (size_t)row * CCH;
      hr[ 0 + l16] = acc0[j];
      hr[16 + l16] = acc1[j];
      hr[32 + l16] = acc2[j];
      hr[48 + l16] = acc3[j];
    }
  }
}

// ---------------------------------------------------------------- 5) agg init: g = h*dinv^2 + b   (self loop + bias)
__global__ void k_agg_init(const float* __restrict__ h,
                           const float* __restrict__ dinv,
                           const float* __restrict__ b,
                           float* __restrict__ g, int n) {
  size_t t = (size_t)blockIdx.x * blockDim.x + threadIdx.x;
  size_t total = (size_t)n * CCH;
  if (t >= total) return;
  int row = (int)(t >> 6);
  int c = (int)(t & 63);
  float di = dinv[row];
  g[t] = h[t] * di * di + b[c];
}

// ---------------------------------------------------------------- 6) edge scatter: g[dst] += h[src]*norm
// thread = (edge, channel): 64 consecutive threads share an edge, so both the
// gather of h[src,:] and the atomic adds to g[dst,:] are fully coalesced and
// stay resident in the 192 MB L2.
__global__ void k_agg_edges(const int* __restrict__ src,
                            const int* __restrict__ dst,
                            const float* __restrict__ h,
                            const float* __restrict__ dinv,
                            float* __restrict__ g, int nE) {
  size_t t = (size_t)blockIdx.x * blockDim.x + threadIdx.x;
  size_t total = (size_t)nE * CCH;
  if (t >= total) return;
  int e = (int)(t >> 6);
  int c = (int)(t & 63);
  int s = src[e];
  int d = dst[e];
  float norm = dinv[s] * dinv[d];
  atomicAdd(&g[(size_t)d * CCH + c], h[(size_t)s * CCH + c] * norm);
}

// ---------------------------------------------------------------- 7) channel pool (sum + max over nodes)
__global__ void __launch_bounds__(256)
k_pool(const float* __restrict__ g, float* __restrict__ sumbuf,
       float* __restrict__ maxbuf, int n) {
  __shared__ float ss[4][CCH];
  __shared__ float sm[4][CCH];
  const int c = threadIdx.x & 63;
  const int q = threadIdx.x >> 6;  // 0..3
  const int group = (int)((blockIdx.x * 256 + threadIdx.x) >> 6);
  const int stride = (int)((gridDim.x * 256) >> 6);
  float s = 0.0f, m = -INFINITY;
  for (int r = group; r < n; r += stride) {
    float v = g[(size_t)r * CCH + c];
    s += v;
    m = fmaxf(m, v);
  }
  ss[q][c] = s;
  sm[q][c] = m;
  __syncthreads();
  if (q == 0) {
#pragma unroll
    for (int j = 1; j < 4; ++j) {
      s += ss[j][c];
      m = fmaxf(m, sm[j][c]);
    }
    atomicAdd(&sumbuf[c], s);
    atomicMaxF(&maxbuf[c], m);
  }
}

// ---------------------------------------------------------------- 8) channel-attention MLP -> ca[64]
__global__ void k_mlp(const float* __restrict__ sumbuf,
                      const float* __restrict__ maxbuf,
                      const float* __restrict__ w1,   // [C, H] row-major
                      const float* __restrict__ w2,   // [H, C] row-major
                      float* __restrict__ ca, float invN) {
  __shared__ float avg[CCH], mx[CCH], ha[HID], hm[HID];
  int t = threadIdx.x;  // 64 threads
  avg[t] = sumbuf[t] * invN;
  mx[t] = maxbuf[t];
  __syncthreads();
  if (t < HID) {
    float sa = 0.0f, sm = 0.0f;
    for (int c = 0; c < CCH; ++c) {
      float w = w1[c * HID + t];
      sa += avg[c] * w;
      sm += mx[c] * w;
    }
    ha[t] = fmaxf(sa, 0.0f);  // relu
    hm[t] = fmaxf(sm, 0.0f);
  }
  __syncthreads();
  float o = 0.0f;
#pragma unroll
  for (int j = 0; j < HID; ++j) o += (ha[j] + hm[j]) * w2[j * CCH + t];
  ca[t] = sigmoidf_(o);
}

// ---------------------------------------------------------------- 9) epilogue: channel gate, spatial gate, residual, relu (in place)
__global__ void __launch_bounds__(256)
k_final(float* __restrict__ g, const float* __restrict__ ca,
        const float* __restrict__ sw, const float* __restrict__ sb, int n) {
  __shared__ float cas[CCH];
  if (threadIdx.x < CCH) cas[threadIdx.x] = ca[threadIdx.x];
  __syncthreads();
  const int wave = threadIdx.x >> 5;
  const int lane = threadIdx.x & 31;
  const int row = blockIdx.x * 8 + wave;
  if (row >= n) return;
  float* gr = g + (size_t)row * CCH;
  float g0 = gr[lane], g1 = gr[lane + 32];
  float v0 = g0 * cas[lane];
  float v1 = g1 * cas[lane + 32];
  float s = v0 + v1;
  float m = fmaxf(v0, v1);
#pragma unroll
  for (int off = 16; off >= 1; off >>= 1) {
    s += __shfl_xor(s, off, 32);
    m = fmaxf(m, __shfl_xor(m, off, 32));
  }
  float mean = s * (1.0f / CCH);
  float sa = sigmoidf_(mean * sw[0] + m * sw[1] + sb[0]);
  gr[lane]      = fmaxf(v0 * sa + g0, 0.0f);
  gr[lane + 32] = fmaxf(v1 * sa + g1, 0.0f);
}

// ================================================================ launcher
extern "C" void kernel_launch(void* const* d_in, const int* in_sizes, int n_in,
                              void* d_out, int out_size, void* d_ws, size_t ws_size,
                              hipStream_t stream) {
  const float* x   = (const float*)d_in[0];
  const int* eidx  = (const int*)d_in[1];
  const float* W   = (const float*)d_in[2];
  const float* b   = (const float*)d_in[3];
  const float* w1  = (const float*)d_in[4];
  const float* w2  = (const float*)d_in[5];
  const float* sw  = (const float*)d_in[6];
  const float* sb  = (const float*)d_in[7];

  const int N = in_sizes[0] / CCH;
  const int E = in_sizes[1] / 2;
  const int* src = eidx;
  const int* dst = eidx + E;

  // workspace layout
  float* h      = (float*)d_ws;              // N*C
  float* dinv   = h + (size_t)N * CCH;       // N   (deg -> dinv in place)
  float* sumbuf = dinv + N;                  // 64
  float* maxbuf = sumbuf + CCH;              // 64
  float* ca     = maxbuf + CCH;              // 64
  float* g      = (float*)d_out;             // N*C (agg, then final output in place)

  const int TB = 256;

  // 1) init
  k_init<<<(N + TB - 1) / TB, TB, 0, stream>>>(dinv, sumbuf, maxbuf, N);
  // 2) degree
  k_degree<<<(E + TB - 1) / TB, TB, 0, stream>>>(dst, dinv, E);
  // 3) dinv
  k_dinv<<<(N + TB - 1) / TB, TB, 0, stream>>>(dinv, N);
  // 4) h = x @ W via f32 WMMA
  k_gemm<<<(N + 127) / 128, TB, 0, stream>>>(x, W, h, N);
  // 5) g = h*dinv^2 + b
  {
    size_t total = (size_t)N * CCH;
    k_agg_init<<<(unsigned)((total + TB - 1) / TB), TB, 0, stream>>>(h, dinv, b, g, N);
  }
  // 6) g[dst] += h[src] * dinv[src]*dinv[dst]
  {
    size_t total = (size_t)E * CCH;
    k_agg_edges<<<(unsigned)((total + TB - 1) / TB), TB, 0, stream>>>(src, dst, h, dinv, g, E);
  }
  // 7) channel pooling
  k_pool<<<1024, TB, 0, stream>>>(g, sumbuf, maxbuf, N);
  // 8) channel-attention MLP
  k_mlp<<<1, CCH, 0, stream>>>(sumbuf, maxbuf, w1, w2, ca, 1.0f / (float)N);
  // 9) fused CBAM-spatial + residual + relu
  k_final<<<(N + 7) / 8, TB, 0, stream>>>(g, ca, sw, sb, N);
}